// RecurrentGCN_44341242364543
// MI455X (gfx1250) — compile-verified
//
#include <hip/hip_runtime.h>

// RecurrentGCN (HeteroGCLSTM x2 + linear) for MI455X / gfx1250.
//
// Key algebraic reduction: both _gclstm calls zero-init their hidden states,
// and every SAGEConv runs on those zero states -> each conv term is just its
// bias. Edges and location features never influence the patient output.
// The model collapses to 3 dense stages (see analysis). We use the native
// fp32 WMMA (v_wmma_f32_16x16x4_f32) so precision matches the fp32 reference
// while still driving the matrix pipes. Problem is HBM-bound (~5 GFLOP,
// ~0.1 GB traffic @ 23.3 TB/s).

typedef __attribute__((ext_vector_type(2))) float v2f;
typedef __attribute__((ext_vector_type(8))) float v8f;

__device__ __forceinline__ float sigmoid_f(float x) {
  return 1.0f / (1.0f + expf(-x));
}

// One fused GCLSTM gate layer: H[r,c] = relu( sig(o)*tanh( sig(i)*tanh(c) ) )
// with gate g = X @ Wg + (bga + bgb)  (bias = b_p + lp_bl, both length N).
// X: [rows, K] row-major, Wg: [K, N] row-major, H: [rows, N].
// Block: 16 rows, N/16 waves (one 16-col WMMA tile per wave), blockDim = N*2.
template<int K, int N>
__global__ __launch_bounds__(N * 2)
void gclstm_layer_kernel(const float* __restrict__ X,
                         const float* __restrict__ Wi,
                         const float* __restrict__ Wc,
                         const float* __restrict__ Wo,
                         const float* __restrict__ bia, const float* __restrict__ bib,
                         const float* __restrict__ bca, const float* __restrict__ bcb,
                         const float* __restrict__ boa, const float* __restrict__ bob,
                         float* __restrict__ H, int rows) {
  constexpr int LDSS = K + 4;  // pad: bank stride 4 -> conflict-free A reads
  __shared__ float Xs[16][LDSS];

  const int row0 = blockIdx.x * 16;
  const int tid = threadIdx.x;

  // Cooperative stage of the 16 x K A-tile into LDS.
  for (int t = tid; t < 16 * K; t += N * 2) {
    const int r = t / K, c = t % K;
    int rg = row0 + r;
    if (rg > rows - 1) rg = rows - 1;  // clamp (rows%16==0 in practice)
    Xs[r][c] = X[(size_t)rg * K + c];
  }
  __syncthreads();

  const int lane = tid & 31;
  const int wave = tid >> 5;   // 0 .. N/16-1
  const int half = lane >> 4;  // 0/1: K-split for A/B, M-split for C/D
  const int idx  = lane & 15;  // M for A, N for B/C/D
  const int col  = wave * 16 + idx;

  const float* wi = Wi + col;
  const float* wc = Wc + col;
  const float* wo = Wo + col;

  v8f ai = {}; v8f ac = {}; v8f ao = {};
#pragma unroll
  for (int kk = 0; kk < K / 4; ++kk) {
    const int kb = kk * 4 + half * 2;
    // A fragment (16x4 f32): lane half selects K pair {kb, kb+1}, M = idx.
    v2f a;  a.x  = Xs[idx][kb];           a.y  = Xs[idx][kb + 1];
    // B fragments (4x16 f32): N = idx across lanes, K pair per half.
    v2f bi; bi.x = wi[(size_t)kb * N];    bi.y = wi[(size_t)(kb + 1) * N];
    v2f bc; bc.x = wc[(size_t)kb * N];    bc.y = wc[(size_t)(kb + 1) * N];
    v2f bo; bo.x = wo[(size_t)kb * N];    bo.y = wo[(size_t)(kb + 1) * N];
    // 8 args: (neg_a, A, neg_b, B, c_mod, C, reuse_a, reuse_b)
    ai = __builtin_amdgcn_wmma_f32_16x16x4_f32(false, a, false, bi, (short)0, ai, false, false);
    ac = __builtin_amdgcn_wmma_f32_16x16x4_f32(false, a, false, bc, (short)0, ac, false, false);
    ao = __builtin_amdgcn_wmma_f32_16x16x4_f32(false, a, false, bo, (short)0, ao, false, false);
  }

  // Per-column gate biases: b_p[col] + lp_bl[col].
  const float Bi = bia[col] + bib[col];
  const float Bc = bca[col] + bcb[col];
  const float Bo = boa[col] + bob[col];

  // D layout: VGPR j -> row (j + 8*half), col = idx.
#pragma unroll
  for (int j = 0; j < 8; ++j) {
    const int row = row0 + j + half * 8;
    const float gi = sigmoid_f(ai[j] + Bi);
    const float gc = tanhf(ac[j] + Bc);
    const float go = sigmoid_f(ao[j] + Bo);
    float h = go * tanhf(gi * gc);  // c = i*tanh(.), h = o*tanh(c)
    h = fmaxf(h, 0.0f);             // relu fused (both layers feed a relu)
    if (row < rows) H[(size_t)row * N + col] = h;
  }
}

// Out[row, 0:6] = H2r[row, 0:64] @ lin_W[64,6] + lin_b[6]   (H2r already relu'd)
__global__ __launch_bounds__(256)
void final_linear_kernel(const float* __restrict__ H,
                         const float* __restrict__ W,
                         const float* __restrict__ b,
                         float* __restrict__ out, int rows) {
  const int row = blockIdx.x * blockDim.x + threadIdx.x;
  if (row >= rows) return;
  float acc[6];
#pragma unroll
  for (int j = 0; j < 6; ++j) acc[j] = b[j];
  const float* h = H + (size_t)row * 64;
#pragma unroll
  for (int k = 0; k < 64; ++k) {
    const float hv = h[k];
#pragma unroll
    for (int j = 0; j < 6; ++j) acc[j] = fmaf(hv, W[k * 6 + j], acc[j]);
  }
#pragma unroll
  for (int j = 0; j < 6; ++j) out[(size_t)row * 6 + j] = acc[j];
}

extern "C" void kernel_launch(void* const* d_in, const int* in_sizes, int n_in,
                              void* d_out, int out_size, void* d_ws, size_t ws_size,
                              hipStream_t stream) {
  (void)n_in; (void)out_size; (void)ws_size;

  // Detect flattening order of the nested inputs dict:
  //  - insertion order: d_in[0] = x_patient   (50000*64 = 3,200,000 elems)
  //  - jax sorted-pytree order: d_in[0] = edge_index_lp (2*800000 = 1,600,000)
  const bool sorted = (in_sizes[0] == 1600000);

  int xp_i, l1_b, l2_b, linW_i, linB_i;  // base indices
  int fWp, fbp, flpbl;                   // field offsets within a gate (10 leaves)
  int gI, gC, gO;                        // gate slot (x10) within a layer
  if (sorted) {
    // order: edge_lp, edge_pl, params{l1,l2,lin_W,lin_b}, x_location, x_patient
    // layer fields sorted: W_l,W_p,b_l,b_p,lp_Wl,lp_Wr,lp_bl,pl_Wl,pl_Wr,pl_bl
    // gates sorted: c,f,i,o
    xp_i = 85; l1_b = 2; l2_b = 42; linW_i = 82; linB_i = 83;
    fWp = 1; fbp = 3; flpbl = 6;
    gC = 0; gI = 2; gO = 3;
  } else {
    // insertion: x_patient, x_location, edge_pl, edge_lp, l1(40), l2(40), lin_W, lin_b
    // layer fields: W_p,W_l,b_p,b_l,pl_Wl,pl_Wr,pl_bl,lp_Wl,lp_Wr,lp_bl
    // gates: i,f,c,o
    xp_i = 0; l1_b = 4; l2_b = 44; linW_i = 84; linB_i = 85;
    fWp = 0; fbp = 2; flpbl = 9;
    gI = 0; gC = 2; gO = 3;
  }

  auto F = [&](int i) { return (const float*)d_in[i]; };

  const float* X = F(xp_i);
  const int rows = in_sizes[xp_i] / 64;  // 50000

  // Layer 1 (K=64 -> N=128): weights W_p and bias pair (b_p, lp_bl) per gate.
  const float* W1i = F(l1_b + gI * 10 + fWp);
  const float* W1c = F(l1_b + gC * 10 + fWp);
  const float* W1o = F(l1_b + gO * 10 + fWp);
  const float* b1ia = F(l1_b + gI * 10 + fbp), *b1ib = F(l1_b + gI * 10 + flpbl);
  const float* b1ca = F(l1_b + gC * 10 + fbp), *b1cb = F(l1_b + gC * 10 + flpbl);
  const float* b1oa = F(l1_b + gO * 10 + fbp), *b1ob = F(l1_b + gO * 10 + flpbl);

  // Layer 2 (K=128 -> N=64)
  const float* W2i = F(l2_b + gI * 10 + fWp);
  const float* W2c = F(l2_b + gC * 10 + fWp);
  const float* W2o = F(l2_b + gO * 10 + fWp);
  const float* b2ia = F(l2_b + gI * 10 + fbp), *b2ib = F(l2_b + gI * 10 + flpbl);
  const float* b2ca = F(l2_b + gC * 10 + fbp), *b2cb = F(l2_b + gC * 10 + flpbl);
  const float* b2oa = F(l2_b + gO * 10 + fbp), *b2ob = F(l2_b + gO * 10 + flpbl);

  const float* linW = F(linW_i);
  const float* linB = F(linB_i);

  // Workspace: H1 [rows,128] then H2r [rows,64]
  float* H1 = (float*)d_ws;
  float* H2 = H1 + (size_t)rows * 128;

  const int nblk = (rows + 15) / 16;  // 3125 (rows % 16 == 0)

  gclstm_layer_kernel<64, 128><<<nblk, 256, 0, stream>>>(
      X, W1i, W1c, W1o, b1ia, b1ib, b1ca, b1cb, b1oa, b1ob, H1, rows);

  gclstm_layer_kernel<128, 64><<<nblk, 128, 0, stream>>>(
      H1, W2i, W2c, W2o, b2ia, b2ib, b2ca, b2cb, b2oa, b2ob, H2, rows);

  final_linear_kernel<<<(rows + 255) / 256, 256, 0, stream>>>(
      H2, linW, linB, (float*)d_out, rows);
}